// GCNRegress_66675072303277
// MI455X (gfx1250) — compile-verified
//
#include <hip/hip_runtime.h>
#include <hip/hip_bf16.h>

typedef __attribute__((ext_vector_type(2))) float v2f;
typedef __attribute__((ext_vector_type(8))) float v8f;
typedef long long ll;

// ---------------------------------------------------------------------------
// Degree / normalization precompute
// ---------------------------------------------------------------------------
__global__ void init_deg(float* __restrict__ deg, int n) {
    int i = blockIdx.x * blockDim.x + threadIdx.x;
    if (i < n) deg[i] = 1.0f;  // self-loop weight
}

__global__ void deg_edges(const ll* __restrict__ col, const float* __restrict__ ew,
                          float* __restrict__ deg, int E) {
    int e = blockIdx.x * blockDim.x + threadIdx.x;
    if (e < E) atomicAdd(&deg[(int)col[e]], ew[e]);
}

__global__ void node_norm(const float* __restrict__ deg, float* __restrict__ dinv,
                          float* __restrict__ snorm, int n) {
    int i = blockIdx.x * blockDim.x + threadIdx.x;
    if (i < n) {
        float d = deg[i];
        float r = (d > 0.0f) ? rsqrtf(d) : 0.0f;
        dinv[i] = r;
        snorm[i] = r * r;
    }
}

// ---------------------------------------------------------------------------
// CSR-by-destination build (once per call; ~1M cheap atomics total, replaces
// ~250M f32 atomics in the per-edge scatter aggregation)
// ---------------------------------------------------------------------------
__global__ void zero_i32(int* __restrict__ p, int n) {
    int i = blockIdx.x * blockDim.x + threadIdx.x;
    if (i < n) p[i] = 0;
}

__global__ void hist_edges(const ll* __restrict__ col, int* __restrict__ cnt, int E) {
    int e = blockIdx.x * blockDim.x + threadIdx.x;
    if (e < E) atomicAdd(&cnt[(int)col[e]], 1);
}

// Single-block exclusive scan: cnt[0..n) -> rowptr[0..n], n <= 1024*16
__global__ void __launch_bounds__(1024)
scan_excl(const int* __restrict__ cnt, int* __restrict__ rowptr, int n) {
    __shared__ int sums[1024];
    int tid = threadIdx.x;
    int per = (n + 1023) / 1024;
    int start = tid * per;
    int local = 0;
    for (int i = 0; i < per; ++i) {
        int idx = start + i;
        if (idx < n) local += cnt[idx];
    }
    sums[tid] = local;
    __syncthreads();
    for (int off = 1; off < 1024; off <<= 1) {
        int v = (tid >= off) ? sums[tid - off] : 0;
        __syncthreads();
        if (tid >= off) sums[tid] += v;
        __syncthreads();
    }
    int carry = (tid == 0) ? 0 : sums[tid - 1];
    for (int i = 0; i < per; ++i) {
        int idx = start + i;
        if (idx < n) { rowptr[idx] = carry; carry += cnt[idx]; }
    }
    if (tid == 0) rowptr[n] = sums[1023];
}

__global__ void copy_i32(const int* __restrict__ a, int* __restrict__ b, int n) {
    int i = blockIdx.x * blockDim.x + threadIdx.x;
    if (i < n) b[i] = a[i];
}

// Scatter edges into CSR slots; also computes the per-edge norm weight.
__global__ void fill_csr(const ll* __restrict__ row, const ll* __restrict__ col,
                         const float* __restrict__ ew, const float* __restrict__ dinv,
                         int* __restrict__ cursor, int* __restrict__ srcs,
                         float* __restrict__ wts, int E) {
    int e = blockIdx.x * blockDim.x + threadIdx.x;
    if (e >= E) return;
    int r = (int)row[e], c = (int)col[e];
    int p = atomicAdd(&cursor[c], 1);
    srcs[p] = r;
    wts[p] = dinv[r] * ew[e] * dinv[c];
}

// W[K,N] row-major -> Wt[N,K] (coalesced read; tiny, L2-resident)
__global__ void transpose_w(const float* __restrict__ W, float* __restrict__ Wt,
                            int K, int N) {
    int idx = blockIdx.x * blockDim.x + threadIdx.x;
    if (idx >= K * N) return;
    int k = idx / N, nn = idx % N;
    Wt[(size_t)nn * K + k] = W[idx];
}

// ---------------------------------------------------------------------------
// CSR aggregation: agg[i] = snorm[i]*h[i] + sum_{e in CSR[i]} w_e * h[src_e]
// One block per node, F/4 float4 lanes; neighbor (src,w) pairs staged through
// LDS in chunks of 64 so per-neighbor metadata reads are LDS broadcasts.
// Zero atomics; plain b128 stores.
// ---------------------------------------------------------------------------
#define CSR_CHUNK 64
__global__ void agg_csr_f4(const float* __restrict__ h, const int* __restrict__ rowptr,
                           const int* __restrict__ srcs, const float* __restrict__ wts,
                           const float* __restrict__ snorm, float* __restrict__ agg, int F) {
    __shared__ int   s_src[CSR_CHUNK];
    __shared__ float s_w[CSR_CHUNK];
    const int node = blockIdx.x;
    const int f = threadIdx.x;  // 0..F/4-1 (>= 64 for F in {256,512})
    const int s = rowptr[node], e_end = rowptr[node + 1];

    float sn = snorm[node];
    float4 hv = ((const float4*)(h + (size_t)node * F))[f];
    float4 acc;
    acc.x = sn * hv.x; acc.y = sn * hv.y; acc.z = sn * hv.z; acc.w = sn * hv.w;

    for (int base = s; base < e_end; base += CSR_CHUNK) {
        int m = min(CSR_CHUNK, e_end - base);
        if (f < m) { s_src[f] = srcs[base + f]; s_w[f] = wts[base + f]; }
        __syncthreads();
        for (int j = 0; j < m; ++j) {
            float w = s_w[j];
            float4 v = ((const float4*)(h + (size_t)s_src[j] * F))[f];
            acc.x = fmaf(w, v.x, acc.x);
            acc.y = fmaf(w, v.y, acc.y);
            acc.z = fmaf(w, v.z, acc.z);
            acc.w = fmaf(w, v.w, acc.w);
        }
        __syncthreads();
    }
    ((float4*)(agg + (size_t)node * F))[f] = acc;
}

// Narrow (F==8) CSR aggregation for the last layer, bias fused; writes d_out.
__global__ void out_csr8(const float* __restrict__ t, const int* __restrict__ rowptr,
                         const int* __restrict__ srcs, const float* __restrict__ wts,
                         const float* __restrict__ snorm, const float* __restrict__ b,
                         float* __restrict__ out, int n) {
    int tid = blockIdx.x * blockDim.x + threadIdx.x;
    int i = tid >> 3, j = tid & 7;
    if (i >= n) return;
    float acc = snorm[i] * t[i * 8 + j] + b[j];
    int s = rowptr[i], e = rowptr[i + 1];
    for (int p = s; p < e; ++p) acc = fmaf(wts[p], t[(size_t)srcs[p] * 8 + j], acc);
    out[i * 8 + j] = acc;
}

// ---------------------------------------------------------------------------
// WMMA fp32 GEMM: C[M,N] = A[M,K] * Bt^T[K,N] + bias (+ReLU), Bt is [N,K].
// One wave32 computes a 16x64 tile (4 N-subtiles register-blocked),
// software-pipelined K loop of V_WMMA_F32_16X16X4_F32.
// ---------------------------------------------------------------------------
__global__ void __launch_bounds__(32)
gemm_wmma_f32_x4(const float* __restrict__ A, const float* __restrict__ Bt,
                 const float* __restrict__ bias, float* __restrict__ C,
                 int K, int N, int relu) {
    const int m0 = blockIdx.x << 4;
    const int n0 = blockIdx.y << 6;
    const int t = threadIdx.x;
    const int half = t >> 4;
    const int l16 = t & 15;

    const float* ap = A + (size_t)(m0 + l16) * K + 2 * half;
    const float* bp = Bt + (size_t)(n0 + l16) * K + 2 * half;
    const size_t bs = (size_t)16 * K;

    v8f acc0 = {}, acc1 = {}, acc2 = {}, acc3 = {};

    v2f a  = *(const v2f*)(ap);
    v2f b0 = *(const v2f*)(bp);
    v2f b1 = *(const v2f*)(bp + bs);
    v2f b2 = *(const v2f*)(bp + 2 * bs);
    v2f b3 = *(const v2f*)(bp + 3 * bs);

    for (int k = 4; k < K; k += 4) {
        v2f an  = *(const v2f*)(ap + k);
        v2f bn0 = *(const v2f*)(bp + k);
        v2f bn1 = *(const v2f*)(bp + bs + k);
        v2f bn2 = *(const v2f*)(bp + 2 * bs + k);
        v2f bn3 = *(const v2f*)(bp + 3 * bs + k);
        acc0 = __builtin_amdgcn_wmma_f32_16x16x4_f32(false, a, false, b0, (short)0, acc0, false, false);
        acc1 = __builtin_amdgcn_wmma_f32_16x16x4_f32(false, a, false, b1, (short)0, acc1, false, false);
        acc2 = __builtin_amdgcn_wmma_f32_16x16x4_f32(false, a, false, b2, (short)0, acc2, false, false);
        acc3 = __builtin_amdgcn_wmma_f32_16x16x4_f32(false, a, false, b3, (short)0, acc3, false, false);
        a = an; b0 = bn0; b1 = bn1; b2 = bn2; b3 = bn3;
    }
    acc0 = __builtin_amdgcn_wmma_f32_16x16x4_f32(false, a, false, b0, (short)0, acc0, false, false);
    acc1 = __builtin_amdgcn_wmma_f32_16x16x4_f32(false, a, false, b1, (short)0, acc1, false, false);
    acc2 = __builtin_amdgcn_wmma_f32_16x16x4_f32(false, a, false, b2, (short)0, acc2, false, false);
    acc3 = __builtin_amdgcn_wmma_f32_16x16x4_f32(false, a, false, b3, (short)0, acc3, false, false);

    v8f accs[4] = {acc0, acc1, acc2, acc3};
    #pragma unroll
    for (int j = 0; j < 4; ++j) {
        int nc = n0 + 16 * j + l16;
        float bv = bias[nc];
        float* crow = C + (size_t)(m0 + 8 * half) * N + nc;
        #pragma unroll
        for (int r = 0; r < 8; ++r) {
            float v = accs[j][r] + bv;
            if (relu) v = fmaxf(v, 0.0f);
            crow[(size_t)r * N] = v;
        }
    }
}

// Tiny GEMM for layer 3 (Fout = 8): thread per (node, j).
__global__ void gemm_small8(const float* __restrict__ h, const float* __restrict__ W,
                            float* __restrict__ t, int M, int K) {
    int tid = blockIdx.x * blockDim.x + threadIdx.x;
    int i = tid >> 3, j = tid & 7;
    if (i >= M) return;
    const float* hr = h + (size_t)i * K;
    float acc = 0.0f;
    for (int k = 0; k < K; ++k) acc = fmaf(hr[k], W[k * 8 + j], acc);
    t[(size_t)i * 8 + j] = acc;
}

// ---------------------------------------------------------------------------
static inline size_t rup4(size_t x) { return (x + 3) & ~(size_t)3; }

extern "C" void kernel_launch(void* const* d_in, const int* in_sizes, int n_in,
                              void* d_out, int out_size, void* d_ws, size_t ws_size,
                              hipStream_t stream) {
    const float* x   = (const float*)d_in[0];
    const ll*    ei  = (const ll*)d_in[1];
    const float* ew  = (const float*)d_in[2];
    const float* W1  = (const float*)d_in[3];
    const float* b1  = (const float*)d_in[4];
    const float* W2  = (const float*)d_in[5];
    const float* b2  = (const float*)d_in[6];
    const float* W3  = (const float*)d_in[7];
    const float* b3  = (const float*)d_in[8];
    float* out = (float*)d_out;

    const int F0 = 256;
    const int n  = in_sizes[0] / F0;      // 10000
    const int E  = in_sizes[2];           // 320000
    const int F1 = in_sizes[4];           // 512
    const int F2 = in_sizes[6];           // 768
    const ll* row = ei;                   // edge_index[0] (source)
    const ll* col = ei + E;               // edge_index[1] (target)

    // Workspace layout in 4-byte words, 16B-aligned regions.
    char* wsb = (char*)d_ws;
    size_t off = 0;
    auto alloc = [&](size_t words) { void* p = wsb + off * 4; off += rup4(words); return p; };
    float* deg    = (float*)alloc(n);
    float* dinv   = (float*)alloc(n);
    float* snorm  = (float*)alloc(n);
    int*   cnt    = (int*)alloc(n);
    int*   rowptr = (int*)alloc(n + 1);
    int*   cursor = (int*)alloc(n);
    int*   srcs   = (int*)alloc(E);
    float* wts    = (float*)alloc(E);
    float* t3     = (float*)alloc((size_t)n * 8);
    float* Wt1    = (float*)alloc((size_t)F0 * F1);
    float* Wt2    = (float*)alloc((size_t)F1 * F2);
    float* bufA   = (float*)alloc((size_t)n * F1);   // agg buffer (max width 512)
    float* bufB   = (float*)alloc((size_t)n * F2);   // hidden buffer (max width 768)

    const int TB = 256;
    const int nb_n = (n + TB - 1) / TB;
    const int nb_e = (E + TB - 1) / TB;

    // --- degree + symmetric normalization (shared by all 3 layers) ---
    init_deg<<<nb_n, TB, 0, stream>>>(deg, n);
    deg_edges<<<nb_e, TB, 0, stream>>>(col, ew, deg, E);
    node_norm<<<nb_n, TB, 0, stream>>>(deg, dinv, snorm, n);

    // --- CSR-by-destination build ---
    zero_i32<<<nb_n, TB, 0, stream>>>(cnt, n);
    hist_edges<<<nb_e, TB, 0, stream>>>(col, cnt, E);
    scan_excl<<<1, 1024, 0, stream>>>(cnt, rowptr, n);
    copy_i32<<<nb_n, TB, 0, stream>>>(rowptr, cursor, n);
    fill_csr<<<nb_e, TB, 0, stream>>>(row, col, ew, dinv, cursor, srcs, wts, E);

    // --- weight transposes for contiguous WMMA B-fragment loads ---
    transpose_w<<<(F0 * F1 + TB - 1) / TB, TB, 0, stream>>>(W1, Wt1, F0, F1);
    transpose_w<<<(F1 * F2 + TB - 1) / TB, TB, 0, stream>>>(W2, Wt2, F1, F2);

    // --- layer 1: agg(x) @ W1 + b1, ReLU ---  (aggregate at width 256)
    {
        agg_csr_f4<<<n, F0 / 4, 0, stream>>>(x, rowptr, srcs, wts, snorm, bufA, F0);
        dim3 g(n / 16, F1 / 64);
        gemm_wmma_f32_x4<<<g, 32, 0, stream>>>(bufA, Wt1, b1, bufB, F0, F1, /*relu=*/1);
    }

    // --- layer 2: agg(h1) @ W2 + b2, ReLU ---  (aggregate at width 512)
    {
        agg_csr_f4<<<n, F1 / 4, 0, stream>>>(bufB, rowptr, srcs, wts, snorm, bufA, F1);
        dim3 g(n / 16, F2 / 64);
        gemm_wmma_f32_x4<<<g, 32, 0, stream>>>(bufA, Wt2, b2, bufB, F1, F2, /*relu=*/1);
    }

    // --- layer 3: agg(h2 @ W3) + b3 ---  (transform first, aggregate at width 8)
    {
        int tot = n * 8;
        gemm_small8<<<(tot + TB - 1) / TB, TB, 0, stream>>>(bufB, W3, t3, n, F2);
        out_csr8<<<(tot + TB - 1) / TB, TB, 0, stream>>>(t3, rowptr, srcs, wts, snorm, b3, out, n);
    }
}